// GraphConvolution_41953240547884
// MI455X (gfx1250) — compile-verified
//
#include <hip/hip_runtime.h>

typedef __attribute__((ext_vector_type(2))) float v2f;
typedef __attribute__((ext_vector_type(8))) float v8f;

#define D_FEAT 128   // D_IN == D_OUT == 128

// ---------------------------------------------------------------------------
// Kernel 1: zero the output accumulator (harness poisons d_out with 0xAA).
// ---------------------------------------------------------------------------
__global__ void gcn_zero(float4* __restrict__ out, int n4) {
  int i = blockIdx.x * blockDim.x + threadIdx.x;
  if (i < n4) out[i] = make_float4(0.f, 0.f, 0.f, 0.f);
}

// ---------------------------------------------------------------------------
// Kernel 2: support = X @ W using V_WMMA_F32_16X16X4_F32.
// Block = 256 threads (8 waves). Block owns a 16-row M tile; wave w owns the
// 16-col N tile [16w, 16w+16). W (64KB) and the 16x128 X tile (8KB) are staged
// in LDS once per block; K=128 swept in steps of 4 (32 WMMAs per wave).
// ---------------------------------------------------------------------------
__global__ void gcn_gemm_wmma(const float* __restrict__ X,
                              const float* __restrict__ W,
                              float* __restrict__ support,
                              int n_nodes) {
  extern __shared__ float smem[];
  float* sW = smem;                      // [128][128]
  float* sX = smem + D_FEAT * D_FEAT;    // [16][128]

  const int tid   = threadIdx.x;
  const int wave  = tid >> 5;
  const int lane  = tid & 31;
  const int mbase = blockIdx.x * 16;

  // Stage W cooperatively (4096 float4 / 256 threads = 16 each).
  const float4* W4  = (const float4*)W;
  float4*       sW4 = (float4*)sW;
  #pragma unroll 4
  for (int i = tid; i < (D_FEAT * D_FEAT) / 4; i += 256) sW4[i] = W4[i];

  // Stage the 16x128 X tile (zero-fill guard for a partial last tile).
  const float4* X4  = (const float4*)X;
  float4*       sX4 = (float4*)sX;
  for (int i = tid; i < (16 * D_FEAT) / 4; i += 256) {
    int r  = i / (D_FEAT / 4);
    int cc = i % (D_FEAT / 4);
    float4 v = make_float4(0.f, 0.f, 0.f, 0.f);
    if (mbase + r < n_nodes) v = X4[(size_t)(mbase + r) * (D_FEAT / 4) + cc];
    sX4[i] = v;
  }
  __syncthreads();

  // ISA 7.12.2 f32 fragment layouts (wave32):
  //  A 16x4 : lane<16 -> M=lane,    regs hold K = kk+0, kk+1
  //           lane>=16 -> M=lane-16, regs hold K = kk+2, kk+3
  //  B 4x16 : lane<16 -> N=lane,    regs hold K = kk+0, kk+1
  //           lane>=16 -> N=lane-16, regs hold K = kk+2, kk+3
  const int m    = lane & 15;
  const int koff = (lane >> 4) << 1;     // 0 or 2
  const int n0   = wave * 16;
  const int nn   = n0 + (lane & 15);

  v8f acc = {};
  #pragma unroll
  for (int kk = 0; kk < D_FEAT; kk += 4) {
    v2f a, b;
    a[0] = sX[m * D_FEAT + kk + koff];
    a[1] = sX[m * D_FEAT + kk + koff + 1];
    b[0] = sW[(kk + koff)     * D_FEAT + nn];
    b[1] = sW[(kk + koff + 1) * D_FEAT + nn];
    // (neg_a, A, neg_b, B, c_mod, C, reuse_a, reuse_b)
    acc = __builtin_amdgcn_wmma_f32_16x16x4_f32(
        false, a, false, b, (short)0, acc, false, false);
  }

  // C/D layout: VGPR r -> row M = r + 8*(lane/16), col N = lane%16.
  const int mr0 = (lane >> 4) << 3;
  #pragma unroll
  for (int r = 0; r < 8; ++r) {
    int mm = mbase + mr0 + r;
    if (mm < n_nodes)
      support[(size_t)mm * D_FEAT + n0 + (lane & 15)] = acc[r];
  }
}

// ---------------------------------------------------------------------------
// Kernel 3: out[row[e]] += ew[e] * support[col[e]].
// One wave per edge; each lane moves a float4 (32 lanes * 16B = 512B row) and
// issues 4 f32 atomic adds. support/out are L2-resident (51.2MB each < 192MB).
// ---------------------------------------------------------------------------
__global__ void gcn_scatter(const float* __restrict__ support,
                            const int* __restrict__ row,
                            const int* __restrict__ col,
                            const float* __restrict__ ew,
                            float* __restrict__ out,
                            int n_edges) {
  const int gw   = (blockIdx.x * blockDim.x + threadIdx.x) >> 5;  // edge id
  const int lane = threadIdx.x & 31;
  if (gw >= n_edges) return;

  const int   r = row[gw];
  const int   c = col[gw];
  const float w = ew[gw];

  // Prefetch a future edge's source row into cache (global_prefetch_b8).
  const int pf = gw + 512;
  if (pf < n_edges)
    __builtin_prefetch(support + (size_t)col[pf] * D_FEAT + lane * 4, 0, 1);

  const float4 v = ((const float4*)(support + (size_t)c * D_FEAT))[lane];
  float* dst = out + (size_t)r * D_FEAT + lane * 4;
  atomicAdd(dst + 0, w * v.x);
  atomicAdd(dst + 1, w * v.y);
  atomicAdd(dst + 2, w * v.z);
  atomicAdd(dst + 3, w * v.w);
}

// ---------------------------------------------------------------------------
extern "C" void kernel_launch(void* const* d_in, const int* in_sizes, int n_in,
                              void* d_out, int out_size, void* d_ws, size_t ws_size,
                              hipStream_t stream) {
  const float* X   = (const float*)d_in[0];   // [N, 128] f32
  const int*   row = (const int*)  d_in[1];   // [E] i32
  const int*   col = (const int*)  d_in[2];   // [E] i32
  const float* ew  = (const float*)d_in[3];   // [E] f32
  const float* W   = (const float*)d_in[4];   // [128, 128] f32
  float* out       = (float*)d_out;           // [N, 128] f32
  float* support   = (float*)d_ws;            // [N, 128] f32 scratch

  const int n_nodes = in_sizes[0] / D_FEAT;
  const int n_edges = in_sizes[1];

  // 1) zero output
  const int n4 = out_size / 4;
  gcn_zero<<<(n4 + 255) / 256, 256, 0, stream>>>((float4*)out, n4);

  // 2) support = X @ W (WMMA f32)
  const int mblocks = (n_nodes + 15) / 16;
  const size_t lds_bytes = (size_t)(D_FEAT * D_FEAT + 16 * D_FEAT) * sizeof(float); // 72KB
  gcn_gemm_wmma<<<mblocks, 256, lds_bytes, stream>>>(X, W, support, n_nodes);

  // 3) scatter-add over edges (one wave per edge)
  const long long threads = (long long)n_edges * 32;
  const int sblocks = (int)((threads + 255) / 256);
  gcn_scatter<<<sblocks, 256, 0, stream>>>(support, row, col, ew, out, n_edges);
}